// Filter_feature_22926535426500
// MI455X (gfx1250) — compile-verified
//
#include <hip/hip_runtime.h>
#include <hip/hip_bf16.h>
#include <stdint.h>

// Problem dims (fixed by reference setup_inputs)
#define BB 2048
#define TT 4096
#define NF 32
#define NH 8
#define NP 6
#define BN_EPS 1e-5f

typedef __attribute__((ext_vector_type(2))) float v2f;
typedef __attribute__((ext_vector_type(8))) float v8f;
typedef __attribute__((ext_vector_type(4))) unsigned int v4u;
typedef __attribute__((ext_vector_type(8))) unsigned int v8u;

// ---------------------------------------------------------------------------
// Kernel 1: batchnorm MLP -> per-row scan constants.  One block, 1024 threads.
// GEMMs use v_wmma_f32_16x16x4_f32 (f32 WMMA, K=4 per op).
// Weights are staged zero-padded (N=16) in LDS so WMMA operand fetch is
// unconditional ds_load (no divergent exec-masked global loads).
// ---------------------------------------------------------------------------
__global__ __launch_bounds__(1024) void mlp_params_kernel(
    const float* __restrict__ features,
    const float* __restrict__ bn0_g, const float* __restrict__ bn0_b,
    const float* __restrict__ W1, const float* __restrict__ b1,
    const float* __restrict__ bn1_g, const float* __restrict__ bn1_b,
    const float* __restrict__ W2, const float* __restrict__ b2,
    const float* __restrict__ bn2_g, const float* __restrict__ bn2_b,
    const float* __restrict__ W3, const float* __restrict__ b3,
    const float* __restrict__ pn,
    float* __restrict__ h1, float* __restrict__ h2, float* __restrict__ par)
{
  __shared__ float redS[1024];
  __shared__ float redS2[1024];
  __shared__ float sc0[NF], sh0[NF];
  __shared__ float sc1[NH], sh1l[NH];
  __shared__ float sc2[NH], sh2l[NH];
  __shared__ float Wp1[16][NF];   // W1 zero-padded to 16 output rows
  __shared__ float Wp2[16][NH];   // W2 zero-padded to 16 output rows

  const int tid  = threadIdx.x;
  const int lane = tid & 31;
  const int wave = tid >> 5;

  // ---- Stage zero-padded weights into LDS ----
  if (tid < 16 * NF) {                       // 512 threads
    int n = tid >> 5, k = tid & 31;
    Wp1[n][k] = (n < NH) ? W1[n * NF + k] : 0.f;
  }
  if (tid < 16 * NH) {                       // 128 threads
    int n = tid >> 3, k = tid & 7;
    Wp2[n][k] = (n < NH) ? W2[n * NH + k] : 0.f;
  }

  // ---- Phase A: bn0 statistics over features (B x 32) ----
  {
    int j = tid & 31, chunk = tid >> 5;  // 32 features x 32 chunks of 64 rows
    float s = 0.f, s2 = 0.f;
    for (int r = chunk * 64; r < chunk * 64 + 64; ++r) {
      float x = features[r * NF + j];
      s += x; s2 += x * x;
    }
    redS[chunk * 32 + j] = s;
    redS2[chunk * 32 + j] = s2;
  }
  __syncthreads();
  if (tid < NF) {
    float s = 0.f, s2 = 0.f;
    for (int c = 0; c < 32; ++c) { s += redS[c * 32 + tid]; s2 += redS2[c * 32 + tid]; }
    float m = s / (float)BB;
    float v = s2 / (float)BB - m * m;
    float sc = rsqrtf(v + BN_EPS) * bn0_g[tid];
    sc0[tid] = sc;
    sh0[tid] = bn0_b[tid] - m * sc;
  }
  __syncthreads();

  // ---- Phase B: h1 = bn0(features) @ W1^T + b1 via WMMA f32 16x16x4 ----
  {
    const int hi   = lane >> 4;      // 0 / 1
    const int mrow = lane & 15;      // A-matrix row within tile
    const int n    = lane & 15;      // B/C/D column
    for (int ti = 0; ti < 4; ++ti) {
      int r0 = (wave * 4 + ti) * 16;
      v8f acc = {};
#pragma unroll
      for (int kc = 0; kc < 8; ++kc) {          // K = 32 -> 8 chunks of 4
        int k0 = kc * 4 + 2 * hi;               // A/B VGPR layout: K = 4kc + j + 2*hi
        int row = r0 + mrow;
        float a0 = features[row * NF + k0    ] * sc0[k0    ] + sh0[k0    ];
        float a1 = features[row * NF + k0 + 1] * sc0[k0 + 1] + sh0[k0 + 1];
        v2f A  = {a0, a1};
        v2f Bm = {Wp1[n][k0], Wp1[n][k0 + 1]};  // unconditional LDS load
        acc = __builtin_amdgcn_wmma_f32_16x16x4_f32(
            false, A, false, Bm, (short)0, acc, false, false);
      }
      if (n < NH) {
        float bias = b1[n];
#pragma unroll
        for (int v = 0; v < 8; ++v) {           // D layout: M = v + 8*hi
          int row = r0 + v + 8 * hi;
          h1[row * NH + n] = acc[v] + bias;
        }
      }
    }
  }
  __threadfence();
  __syncthreads();

  // ---- Phase C: bn1 statistics over h1 (B x 8) ----
  {
    int j = tid & 7, chunk = tid >> 3;  // 8 features x 128 chunks of 16 rows
    float s = 0.f, s2 = 0.f;
    for (int r = chunk * 16; r < chunk * 16 + 16; ++r) {
      float x = h1[r * NH + j];
      s += x; s2 += x * x;
    }
    redS[chunk * 8 + j] = s;
    redS2[chunk * 8 + j] = s2;
  }
  __syncthreads();
  if (tid < NH) {
    float s = 0.f, s2 = 0.f;
    for (int c = 0; c < 128; ++c) { s += redS[c * 8 + tid]; s2 += redS2[c * 8 + tid]; }
    float m = s / (float)BB;
    float v = s2 / (float)BB - m * m;
    float sc = rsqrtf(v + BN_EPS) * bn1_g[tid];
    sc1[tid] = sc;
    sh1l[tid] = bn1_b[tid] - m * sc;
  }
  __syncthreads();

  // ---- Phase D: h2 = relu(bn1(h1)) @ W2^T + b2 via WMMA ----
  {
    const int hi   = lane >> 4;
    const int mrow = lane & 15;
    const int n    = lane & 15;
    for (int ti = 0; ti < 4; ++ti) {
      int r0 = (wave * 4 + ti) * 16;
      v8f acc = {};
#pragma unroll
      for (int kc = 0; kc < 2; ++kc) {          // K = 8 -> 2 chunks of 4
        int k0 = kc * 4 + 2 * hi;
        int row = r0 + mrow;
        float a0 = fmaxf(0.f, h1[row * NH + k0    ] * sc1[k0    ] + sh1l[k0    ]);
        float a1 = fmaxf(0.f, h1[row * NH + k0 + 1] * sc1[k0 + 1] + sh1l[k0 + 1]);
        v2f A  = {a0, a1};
        v2f Bm = {Wp2[n][k0], Wp2[n][k0 + 1]};  // unconditional LDS load
        acc = __builtin_amdgcn_wmma_f32_16x16x4_f32(
            false, A, false, Bm, (short)0, acc, false, false);
      }
      if (n < NH) {
        float bias = b2[n];
#pragma unroll
        for (int v = 0; v < 8; ++v) {
          int row = r0 + v + 8 * hi;
          h2[row * NH + n] = acc[v] + bias;
        }
      }
    }
  }
  __threadfence();
  __syncthreads();

  // ---- Phase E: bn2 statistics over h2 ----
  {
    int j = tid & 7, chunk = tid >> 3;
    float s = 0.f, s2 = 0.f;
    for (int r = chunk * 16; r < chunk * 16 + 16; ++r) {
      float x = h2[r * NH + j];
      s += x; s2 += x * x;
    }
    redS[chunk * 8 + j] = s;
    redS2[chunk * 8 + j] = s2;
  }
  __syncthreads();
  if (tid < NH) {
    float s = 0.f, s2 = 0.f;
    for (int c = 0; c < 128; ++c) { s += redS[c * 8 + tid]; s2 += redS2[c * 8 + tid]; }
    float m = s / (float)BB;
    float v = s2 / (float)BB - m * m;
    float sc = rsqrtf(v + BN_EPS) * bn2_g[tid];
    sc2[tid] = sc;
    sh2l[tid] = bn2_b[tid] - m * sc;
  }
  __syncthreads();

  // ---- Phase F: final linear + sigmoid params + scan constants ----
  for (int r = tid; r < BB; r += 1024) {
    float z[NH];
#pragma unroll
    for (int j = 0; j < NH; ++j)
      z[j] = fmaxf(0.f, h2[r * NH + j] * sc2[j] + sh2l[j]);
    float p[NP];
#pragma unroll
    for (int pi = 0; pi < NP; ++pi) {
      float cr = b3[pi];
#pragma unroll
      for (int j = 0; j < NH; ++j) cr += z[j] * W3[pi * NH + j];
      float x = cr + pn[pi];
      p[pi] = 1.f / (1.f + expf(-x));
    }
    float f_start = 0.5f  + 4.5f  * p[0];
    float f_inf   = 0.01f + 0.49f * p[1];
    float f_decay = 2.0f * p[2];
    float f_T     = 2.0f * p[3];
    float w_off   = p[4];
    float w_T     = 0.01f + 0.49f * p[5];

    const float LN01 = -2.302585092994046f;            // ln(0.1)
    float rdec = 0.7f * expf(f_decay * LN01);          // 0.7 * 0.1^f_decay
    float tau  = 10.f * expf(f_T * LN01);              // 10 * 0.1^f_T
    float Lg   = logf(rdec) / ((float)TT * tau);       // per-step log decay
    float Mg   = expf(Lg);
    float LwA  = w_off / w_T;
    float LwB  = 1.f / ((float)TT * w_T);
    float Mw   = expf(-LwB);

    par[r * 8 + 0] = f_inf;
    par[r * 8 + 1] = f_start - f_inf;
    par[r * 8 + 2] = Lg;
    par[r * 8 + 3] = Mg;
    par[r * 8 + 4] = LwA;
    par[r * 8 + 5] = LwB;
    par[r * 8 + 6] = Mw;
    par[r * 8 + 7] = 0.f;
  }
}

// ---------------------------------------------------------------------------
// Kernel 2: time-varying IIR scan. 8 blocks x 256 threads, 1 lane per row.
// X tiles staged into LDS by the Tensor Data Mover (tensor_load_to_lds),
// double buffered, with LDS padding (row stride 17 floats -> no bank conflicts).
// ---------------------------------------------------------------------------
#define RPB    256               // rows per block
#define CCHUNK 16                // time-columns per TDM tile
#define LROW   (CCHUNK + 1)      // padded LDS row stride (pad 1 DWORD / 16 DWORDs)
#define NCHUNK (TT / CCHUNK)

__device__ __forceinline__ void tdm_load_tile(unsigned lds_off, const float* gptr)
{
  unsigned long long ga = (unsigned long long)(uintptr_t)gptr;
  v4u g0;
  g0[0] = 1u;                                         // count=1, gather off
  g0[1] = lds_off;                                    // lds_addr (bytes)
  g0[2] = (unsigned)(ga & 0xffffffffu);               // global_addr[31:0]
  g0[3] = (unsigned)((ga >> 32) & 0x01ffffffu)        // global_addr[56:32]
        | (2u << 30);                                 // type = 2 (image)
  const unsigned td0 = 0x40000000u;                   // huge tensor dims (tile fully in-bounds)
  const unsigned td1 = 0x40000000u;
  v8u g1;
  g1[0] = (2u << 16)        // data_size = 4 bytes
        | (1u << 20)        // pad_enable
        | (3u << 22)        // pad_interval: 16 DWORDs
        | (0u << 25);       // pad_amount: 1 DWORD
  g1[1] = (td0 & 0xffffu) << 16;                            // tensor_dim0[15:0]
  g1[2] = ((td0 >> 16) & 0xffffu) | ((td1 & 0xffffu) << 16);// dim0 hi | dim1 lo
  g1[3] = ((td1 >> 16) & 0xffffu) | ((unsigned)CCHUNK << 16);// dim1 hi | tile_dim0
  g1[4] = (unsigned)RPB;                                    // tile_dim1 (tile_dim2=0)
  g1[5] = (unsigned)TT;                                     // tensor_dim0_stride lo
  g1[6] = 0u;
  g1[7] = 0u;
  asm volatile("tensor_load_to_lds %0, %1" :: "s"(g0), "s"(g1) : "memory");
}

__global__ __launch_bounds__(RPB) void scan_kernel(
    const float* __restrict__ X, const float* __restrict__ Fscalar,
    const float* __restrict__ par, float* __restrict__ out)
{
  __shared__ float tile[2][RPB * LROW];

  const int tid = threadIdx.x;
  const int row = blockIdx.x * RPB + tid;

  const float f_inf = par[row * 8 + 0];
  const float df    = par[row * 8 + 1];
  const float Lg    = par[row * 8 + 2];
  const float Mg    = par[row * 8 + 3];
  const float LwA   = par[row * 8 + 4];
  const float LwB   = par[row * 8 + 5];
  const float Mw    = par[row * 8 + 6];
  const float c     = sqrtf(cbrtf(2.0f) - 1.0f) * Fscalar[0]
                      / 3.14159265358979323846f;

  const float* Xblk = X + (size_t)blockIdx.x * RPB * TT;
  const unsigned lds0 = (unsigned)(uintptr_t)&tile[0][0];
  const unsigned lds1 = (unsigned)(uintptr_t)&tile[1][0];

  float y1 = 0.f, y2 = 0.f, xp = 0.f, acc = 0.f, wsum = 0.f;

  if (tid < 32) tdm_load_tile(lds0, Xblk);              // prefetch chunk 0

  for (int k = 0; k < NCHUNK; ++k) {
    const int buf = k & 1;
    if (tid < 32) {
      if (k + 1 < NCHUNK) {
        tdm_load_tile(buf ? lds0 : lds1, Xblk + (k + 1) * CCHUNK);
        __builtin_amdgcn_s_wait_tensorcnt(1);           // chunk k landed
      } else {
        __builtin_amdgcn_s_wait_tensorcnt(0);
      }
    }
    __syncthreads();

    const float* tp = &tile[buf][tid * LROW];
    // Re-seed incremental exponentials each chunk (accuracy + overflow safety)
    float g = expf(Lg * (float)(k * CCHUNK));
    float E = expf(LwA - LwB * (float)(k * CCHUNK));

#pragma unroll
    for (int j = 0; j < CCHUNK; ++j) {
      const int i = k * CCHUNK + j;
      const float x = tp[j];
      const float f = f_inf + df * g;
      const float a = (f - c) / (f + c);
      const float b = 0.5f * (a + 1.0f);
      const float w = 1.0f / (1.0f + E);
      if (i == 0) {                 // uniform branch (only chunk 0, step 0)
        y1 = x; y2 = x; xp = x;
      } else {
        float y1n = b * (x + xp) - a * y1;
        float y2n = b * (y1n + y1) - a * y2;
        y1 = y1n; y2 = y2n; xp = x;
      }
      acc  += w * y2;
      wsum += w;
      g *= Mg;
      E *= Mw;
    }
    __syncthreads();                // all readers done before buf is reused
  }

  out[row] = acc / wsum;
}

// ---------------------------------------------------------------------------
extern "C" void kernel_launch(void* const* d_in, const int* in_sizes, int n_in,
                              void* d_out, int out_size, void* d_ws, size_t ws_size,
                              hipStream_t stream) {
  const float* X        = (const float*)d_in[0];
  const float* features = (const float*)d_in[1];
  const float* F        = (const float*)d_in[2];
  const float* bn0_g    = (const float*)d_in[3];
  const float* bn0_b    = (const float*)d_in[4];
  const float* W1       = (const float*)d_in[5];
  const float* b1       = (const float*)d_in[6];
  const float* bn1_g    = (const float*)d_in[7];
  const float* bn1_b    = (const float*)d_in[8];
  const float* W2       = (const float*)d_in[9];
  const float* b2       = (const float*)d_in[10];
  const float* bn2_g    = (const float*)d_in[11];
  const float* bn2_b    = (const float*)d_in[12];
  const float* W3       = (const float*)d_in[13];
  const float* b3       = (const float*)d_in[14];
  const float* pn       = (const float*)d_in[15];

  float* ws  = (float*)d_ws;
  float* h1  = ws;                 // 2048 x 8
  float* h2  = ws + BB * NH;       // 2048 x 8
  float* par = ws + 2 * BB * NH;   // 2048 x 8

  mlp_params_kernel<<<1, 1024, 0, stream>>>(
      features, bn0_g, bn0_b, W1, b1, bn1_g, bn1_b,
      W2, b2, bn2_g, bn2_b, W3, b3, pn, h1, h2, par);

  scan_kernel<<<BB / RPB, RPB, 0, stream>>>(X, F, par, (float*)d_out);
}